// ChannelAttention_36696200577250
// MI455X (gfx1250) — compile-verified
//
#include <hip/hip_runtime.h>
#include <math.h>

#define BATCH 32
#define SEQ   2048
#define CH    1280
#define HID   160
#define NSEG  32
#define ROWS_PER_SEG (SEQ / NSEG)   // 64 rows per partial-reduction segment

typedef __attribute__((ext_vector_type(2))) float v2f;
typedef __attribute__((ext_vector_type(4))) float v4f;
typedef __attribute__((ext_vector_type(8))) float v8f;

// ---------------------------------------------------------------------------
// Pass 1: partial mean/max pooling over the sequence dimension.
// Grid (NSEG, BATCH), 320 threads (10 waves). Each thread owns 4 consecutive
// channels (float4) -> fully coalesced global_load_b128, stride CH per row.
// ---------------------------------------------------------------------------
__global__ void pool_partial_kernel(const float* __restrict__ x,
                                    float* __restrict__ psum,
                                    float* __restrict__ pmax) {
  const int seg = blockIdx.x;
  const int b   = blockIdx.y;
  const int c   = threadIdx.x * 4;  // 320 threads cover 1280 channels
  const float* p = x + ((size_t)b * SEQ + (size_t)seg * ROWS_PER_SEG) * CH + c;
  v4f s = {0.f, 0.f, 0.f, 0.f};
  v4f m = {-INFINITY, -INFINITY, -INFINITY, -INFINITY};
  #pragma unroll 4
  for (int i = 0; i < ROWS_PER_SEG; ++i) {
    const v4f v = *(const v4f*)p;
    s.x += v.x; s.y += v.y; s.z += v.z; s.w += v.w;
    m.x = fmaxf(m.x, v.x); m.y = fmaxf(m.y, v.y);
    m.z = fmaxf(m.z, v.z); m.w = fmaxf(m.w, v.w);
    p += CH;
  }
  const size_t o = ((size_t)b * NSEG + seg) * CH + c;
  *(v4f*)(psum + o) = s;
  *(v4f*)(pmax + o) = m;
}

// ---------------------------------------------------------------------------
// Pass 2: combine NSEG partials -> gap (mean) and gmp (max). Deterministic
// fixed-order reduction (no float atomics).
// ---------------------------------------------------------------------------
__global__ void pool_combine_kernel(const float* __restrict__ psum,
                                    const float* __restrict__ pmax,
                                    float* __restrict__ gap,
                                    float* __restrict__ gmp) {
  const int idx = blockIdx.x * blockDim.x + threadIdx.x;  // over BATCH*CH/4
  if (idx >= BATCH * (CH / 4)) return;
  const int b = idx / (CH / 4);
  const int c = (idx - b * (CH / 4)) * 4;
  v4f s = {0.f, 0.f, 0.f, 0.f};
  v4f m = {-INFINITY, -INFINITY, -INFINITY, -INFINITY};
  #pragma unroll
  for (int seg = 0; seg < NSEG; ++seg) {
    const size_t o = ((size_t)b * NSEG + seg) * CH + c;
    const v4f vs = *(const v4f*)(psum + o);
    const v4f vm = *(const v4f*)(pmax + o);
    s.x += vs.x; s.y += vs.y; s.z += vs.z; s.w += vs.w;
    m.x = fmaxf(m.x, vm.x); m.y = fmaxf(m.y, vm.y);
    m.z = fmaxf(m.z, vm.z); m.w = fmaxf(m.w, vm.w);
  }
  const float inv = 1.f / (float)SEQ;
  v4f g = {s.x * inv, s.y * inv, s.z * inv, s.w * inv};
  *(v4f*)(gap + (size_t)b * CH + c) = g;
  *(v4f*)(gmp + (size_t)b * CH + c) = m;
}

// ---------------------------------------------------------------------------
// MLP layer 1 via f32 WMMA: stacked [64 x 1280] (rows 0..31 = gap, 32..63 =
// gmp) times w1 [1280 x 160], +b1, relu -> h [64 x 160].
// One wave per 16x16 output tile; grid (HID/16, 64/16). K-loop in steps of 8
// with two accumulators to break the WMMA->WMMA RAW dependency chain.
// A-frag layout (16x4 f32): lane m (0..15) v0/v1 = K0/K1; lanes 16..31 = K2/K3.
// B-frag layout (4x16 f32): mirrored with N = lane&15.
// ---------------------------------------------------------------------------
__global__ void mlp1_kernel(const float* __restrict__ gap,
                            const float* __restrict__ gmp,
                            const float* __restrict__ w1,
                            const float* __restrict__ b1,
                            float* __restrict__ h) {
  const int lane = threadIdx.x;          // 32-thread block, EXEC all ones
  const int nl   = lane & 15;
  const int kh   = (lane >> 4) << 1;     // 0 or 2
  const int m0   = blockIdx.y * 16;
  const int n0   = blockIdx.x * 16;
  const int row  = m0 + nl;              // uniform branch: m0 multiple of 16
  const float* arow = (row < BATCH) ? (gap + (size_t)row * CH)
                                    : (gmp + (size_t)(row - BATCH) * CH);
  const int col = n0 + nl;
  v8f acc0 = {0.f, 0.f, 0.f, 0.f, 0.f, 0.f, 0.f, 0.f};
  v8f acc1 = {0.f, 0.f, 0.f, 0.f, 0.f, 0.f, 0.f, 0.f};
  for (int k0 = 0; k0 < CH; k0 += 8) {
    v2f a0, b0;
    a0.x = arow[k0 + kh];
    a0.y = arow[k0 + kh + 1];
    b0.x = w1[(size_t)(k0 + kh) * HID + col];
    b0.y = w1[(size_t)(k0 + kh + 1) * HID + col];
    acc0 = __builtin_amdgcn_wmma_f32_16x16x4_f32(false, a0, false, b0,
                                                 (short)0, acc0, false, false);
    v2f a1, b1v;
    a1.x = arow[k0 + 4 + kh];
    a1.y = arow[k0 + 5 + kh];
    b1v.x = w1[(size_t)(k0 + 4 + kh) * HID + col];
    b1v.y = w1[(size_t)(k0 + 5 + kh) * HID + col];
    acc1 = __builtin_amdgcn_wmma_f32_16x16x4_f32(false, a1, false, b1v,
                                                 (short)0, acc1, false, false);
  }
  // D layout: acc[j] at lane l -> M = m0 + j + 8*(l>=16), N = n0 + (l&15)
  const float bias  = b1[col];
  const int   mbase = m0 + ((lane >> 4) << 3);
  #pragma unroll
  for (int j = 0; j < 8; ++j) {
    const float v = acc0[j] + acc1[j] + bias;
    h[(size_t)(mbase + j) * HID + col] = fmaxf(v, 0.f);
  }
}

// ---------------------------------------------------------------------------
// MLP layer 2 via f32 WMMA: h [64 x 160] times w2 [160 x 1280], +b2 -> y.
// Grid (CH/16, 64/16) = (80, 4) waves.
// ---------------------------------------------------------------------------
__global__ void mlp2_kernel(const float* __restrict__ h,
                            const float* __restrict__ w2,
                            const float* __restrict__ b2,
                            float* __restrict__ y) {
  const int lane = threadIdx.x;
  const int nl   = lane & 15;
  const int kh   = (lane >> 4) << 1;
  const int m0   = blockIdx.y * 16;
  const int n0   = blockIdx.x * 16;
  const float* arow = h + (size_t)(m0 + nl) * HID;
  const int col = n0 + nl;
  v8f acc0 = {0.f, 0.f, 0.f, 0.f, 0.f, 0.f, 0.f, 0.f};
  v8f acc1 = {0.f, 0.f, 0.f, 0.f, 0.f, 0.f, 0.f, 0.f};
  for (int k0 = 0; k0 < HID; k0 += 8) {
    v2f a0, b0;
    a0.x = arow[k0 + kh];
    a0.y = arow[k0 + kh + 1];
    b0.x = w2[(size_t)(k0 + kh) * CH + col];
    b0.y = w2[(size_t)(k0 + kh + 1) * CH + col];
    acc0 = __builtin_amdgcn_wmma_f32_16x16x4_f32(false, a0, false, b0,
                                                 (short)0, acc0, false, false);
    v2f a1, b1v;
    a1.x = arow[k0 + 4 + kh];
    a1.y = arow[k0 + 5 + kh];
    b1v.x = w2[(size_t)(k0 + 4 + kh) * CH + col];
    b1v.y = w2[(size_t)(k0 + 5 + kh) * CH + col];
    acc1 = __builtin_amdgcn_wmma_f32_16x16x4_f32(false, a1, false, b1v,
                                                 (short)0, acc1, false, false);
  }
  const float bias  = b2[col];
  const int   mbase = m0 + ((lane >> 4) << 3);
  #pragma unroll
  for (int j = 0; j < 8; ++j) {
    y[(size_t)(mbase + j) * CH + col] = acc0[j] + acc1[j] + bias;
  }
}

// ---------------------------------------------------------------------------
// attn[b][c] = sigmoid(y_gap[b][c] + y_gmp[b][c])
// ---------------------------------------------------------------------------
__global__ void attn_kernel(const float* __restrict__ y,
                            float* __restrict__ attn) {
  const int idx = blockIdx.x * blockDim.x + threadIdx.x;  // over BATCH*CH
  if (idx >= BATCH * CH) return;
  const int b = idx / CH;
  const int c = idx - b * CH;
  const float v = y[(size_t)b * CH + c] + y[(size_t)(b + BATCH) * CH + c];
  attn[idx] = 1.f / (1.f + expf(-v));
}

// ---------------------------------------------------------------------------
// out = x * attn (broadcast over seq). float4 per thread. Blocks iterate in
// REVERSE order so the tail of x (still L2-resident after the pool pass) is
// touched first; x loads are last-use non-temporal, out stores non-temporal
// so the output stream does not evict x from L2.
// ---------------------------------------------------------------------------
__global__ void scale_kernel(const float* __restrict__ x,
                             const float* __restrict__ attn,
                             float* __restrict__ out) {
  const size_t nvec = (size_t)BATCH * SEQ * (CH / 4);
  const size_t idx =
      (size_t)(gridDim.x - 1 - blockIdx.x) * blockDim.x + threadIdx.x;
  if (idx >= nvec) return;
  const int    c4 = (int)(idx % (CH / 4));
  const size_t bl = idx / (CH / 4);          // b*SEQ + l
  const int    b  = (int)(bl / SEQ);
  const v4f a = *(const v4f*)(attn + (size_t)b * CH + c4 * 4);
  const v4f v = __builtin_nontemporal_load((const v4f*)(x + idx * 4));
  v4f o;
  o.x = v.x * a.x; o.y = v.y * a.y; o.z = v.z * a.z; o.w = v.w * a.w;
  __builtin_nontemporal_store(o, (v4f*)(out + idx * 4));
}

extern "C" void kernel_launch(void* const* d_in, const int* in_sizes, int n_in,
                              void* d_out, int out_size, void* d_ws,
                              size_t ws_size, hipStream_t stream) {
  const float* x  = (const float*)d_in[0];   // [32, 2048, 1280]
  const float* w1 = (const float*)d_in[1];   // [1280, 160]
  const float* b1 = (const float*)d_in[2];   // [160]
  const float* w2 = (const float*)d_in[3];   // [160, 1280]
  const float* b2 = (const float*)d_in[4];   // [1280]
  float* out = (float*)d_out;

  // Workspace carve-up (~11.3 MB total, all fp32, float4-aligned)
  float* ws   = (float*)d_ws;
  float* psum = ws;                                   // B*NSEG*C
  float* pmax = psum + (size_t)BATCH * NSEG * CH;     // B*NSEG*C
  float* gap  = pmax + (size_t)BATCH * NSEG * CH;     // B*C
  float* gmp  = gap  + (size_t)BATCH * CH;            // B*C
  float* h    = gmp  + (size_t)BATCH * CH;            // 64*H
  float* y    = h    + (size_t)(2 * BATCH) * HID;     // 64*C
  float* attn = y    + (size_t)(2 * BATCH) * CH;      // B*C

  pool_partial_kernel<<<dim3(NSEG, BATCH), 320, 0, stream>>>(x, psum, pmax);
  pool_combine_kernel<<<(BATCH * (CH / 4) + 255) / 256, 256, 0, stream>>>(
      psum, pmax, gap, gmp);
  mlp1_kernel<<<dim3(HID / 16, (2 * BATCH) / 16), 32, 0, stream>>>(
      gap, gmp, w1, b1, h);
  mlp2_kernel<<<dim3(CH / 16, (2 * BATCH) / 16), 32, 0, stream>>>(
      h, w2, b2, y);
  attn_kernel<<<(BATCH * CH + 255) / 256, 256, 0, stream>>>(y, attn);

  const size_t nvec = (size_t)BATCH * SEQ * (CH / 4);  // 20,971,520
  scale_kernel<<<(unsigned)((nvec + 255) / 256), 256, 0, stream>>>(
      x, attn, out);
}